// Ewald_vectorized_84224308674561
// MI455X (gfx1250) — compile-verified
//
#include <hip/hip_runtime.h>
#include <math.h>

typedef float v2f __attribute__((ext_vector_type(2)));
typedef float v8f __attribute__((ext_vector_type(8)));

#define KHEMI 4630          // hemisphere lattice points (first-nonzero > 0), origin excluded
#define TILES 290           // ceil(4630/16) 16-kpoint tiles per batch
#define WAVES_PER_BLOCK 4
#define KSQ_MAX 9.869604401089358f      // (2*pi/DL)^2, DL=2
#define NORM_FACTOR 90.0474f
#define SELF_COEF 0.12698727186848197f  // NQ * 1/(sigma*(2*pi)^1.5), NQ=2, sigma=1

// workspace layout (float offsets)
#define WS_IC   0     // [B*9]  2*pi * inv(cell), row-major
#define WS_VOL  144   // [B]    det(cell)
#define WS_SELF 160   // [B]    sum of q0^2+q1^2 per batch
#define WS_SACC 176   // [B]    sum_k w*|S|^2 accumulator
#define WS_BS   192   // [B+1]  batch start offsets (ints)

__global__ void ewald_prep(const float* __restrict__ q,
                           const float* __restrict__ cell,
                           const int* __restrict__ batch,
                           float* __restrict__ ws, int N, int B) {
  int tid = threadIdx.x;
  float* IC   = ws + WS_IC;
  float* vol  = ws + WS_VOL;
  float* sq   = ws + WS_SELF;
  float* sacc = ws + WS_SACC;
  int*   bs   = (int*)(ws + WS_BS);

  if (tid < B) {
    const float* c = cell + tid * 9;
    float a00=c[0],a01=c[1],a02=c[2],a10=c[3],a11=c[4],a12=c[5],a20=c[6],a21=c[7],a22=c[8];
    float c00 =  a11*a22 - a12*a21;
    float c01 = -(a10*a22 - a12*a20);
    float c02 =  a10*a21 - a11*a20;
    float det = a00*c00 + a01*c01 + a02*c02;
    float id  = 6.283185307179586f / det;            // 2*pi / det
    // IC = 2*pi * inv(cell); inv[i][j] = cof[j][i]/det
    IC[tid*9+0] = c00*id;
    IC[tid*9+1] = (-(a01*a22 - a02*a21))*id;
    IC[tid*9+2] = ( a01*a12 - a02*a11 )*id;
    IC[tid*9+3] = c01*id;
    IC[tid*9+4] = ( a00*a22 - a02*a20 )*id;
    IC[tid*9+5] = (-(a00*a12 - a02*a10))*id;
    IC[tid*9+6] = c02*id;
    IC[tid*9+7] = (-(a00*a21 - a01*a20))*id;
    IC[tid*9+8] = ( a00*a11 - a01*a10 )*id;
    vol[tid]  = det;
    sq[tid]   = 0.0f;
    sacc[tid] = 0.0f;
  }
  if (tid <= B) {  // batch is sorted: start[j] = lower_bound(batch, j)
    int lo = 0, hi = N;
    while (lo < hi) { int mid = (lo + hi) >> 1; if (batch[mid] < tid) lo = mid + 1; else hi = mid; }
    bs[tid] = lo;
  }
  __syncthreads();
  for (int n = tid; n < N; n += blockDim.x) {
    float q0 = q[n*2+0], q1 = q[n*2+1];
    atomicAdd(&sq[batch[n]], q0*q0 + q1*q1);
  }
}

// one 2-atom step: load, phase, fast sincos, WMMA accumulate.
// GUARD=false: caller guarantees atom < a1 (straight-line, no masks).
// GUARD=true : clamp index, zero the q column for out-of-range lanes
//              (zero A-column kills the contribution regardless of B).
template <bool GUARD>
__device__ __forceinline__ v8f ewald_step(const float* __restrict__ q,
                                          const float* __restrict__ r,
                                          int atom, int a1, int kt,
                                          float kx, float ky, float kz, v8f acc) {
  const int at = GUARD ? ((atom < a1) ? atom : (a1 - 1)) : atom;
  float q0 = q[at*2+0], q1 = q[at*2+1];
  float rx = r[at*3+0], ry = r[at*3+1], rz = r[at*3+2];
  float phase = kx*rx + ky*ry + kz*rz;
  if (GUARD && atom >= a1) { q0 = 0.0f; q1 = 0.0f; }
  // phases bounded by |k||r| ~ 65 rad: hardware v_sin/v_cos reduction error
  // (~|x|*2^-24 rad) is below the f32 phase quantization itself.
  float s, c;
  __sincosf(phase, &s, &c);
  // B (4x16): lane half 0 -> rows {cos,sin} of atom0; half 1 -> atom1
  v2f bm; bm.x = c; bm.y = s;
  // A (16x4): rows 0,1 = Re channels (cos cols), rows 2,3 = Im channels (sin cols)
  v2f am;
  am.x = (kt == 0) ? q0 : ((kt == 1) ? q1 : 0.0f);   // even (cos) column
  am.y = (kt == 2) ? q0 : ((kt == 3) ? q1 : 0.0f);   // odd (sin) column
  return __builtin_amdgcn_wmma_f32_16x16x4_f32(false, am, false, bm,
                                               (short)0, acc, false, false);
}

__global__ void __launch_bounds__(WAVES_PER_BLOCK*32)
ewald_main(const float* __restrict__ q, const float* __restrict__ r,
           float* __restrict__ ws, int B) {
  const int lane = threadIdx.x & 31;
  const int wave = threadIdx.x >> 5;
  const int tile = blockIdx.x * WAVES_PER_BLOCK + wave;
  if (tile >= B * TILES) return;               // wave-uniform
  const int b = tile / TILES;
  const int t = tile - b * TILES;

  const float* IC = ws + WS_IC + b * 9;
  const int*   bs = (const int*)(ws + WS_BS);
  const int a0 = bs[b], a1 = bs[b + 1];

  const int kt = lane & 15;                    // k-point within tile == WMMA M row / N col
  const int h  = t * 16 + kt;                  // hemisphere index

  // hemisphere enumeration: (i>0, j, l): 4410 | (0, j>0, l): 210 | (0,0,l>0): 10
  int hi = 0, hj = 0, hl = 0;
  if (h < 4410)       { hi = 1 + h / 441; int rem = h - (hi - 1) * 441;
                        hj = rem / 21 - 10; hl = rem % 21 - 10; }
  else if (h < 4620)  { int h2 = h - 4410; hj = 1 + h2 / 21; hl = h2 % 21 - 10; }
  else if (h < KHEMI) { hl = 1 + (h - 4620); }
  const float ni = (float)hi, nj = (float)hj, nl = (float)hl;

  const float kx = IC[0]*ni + IC[1]*nj + IC[2]*nl;
  const float ky = IC[3]*ni + IC[4]*nj + IC[5]*nl;
  const float kz = IC[6]*ni + IC[7]*nj + IC[8]*nl;
  const float ksq = kx*kx + ky*ky + kz*kz;
  float w = 0.0f;
  if (h < KHEMI && ksq > 0.0f && ksq <= KSQ_MAX)
    w = 2.0f * expf(-0.5f * ksq) / (ksq + 1e-12f);   // factor 2: hemisphere pair

  // cull tiles fully outside the k-cutoff (wave-uniform since halves match)
  if (__ballot(w != 0.0f) == 0ull) return;

  const int half = lane >> 4;                  // selects atom0 / atom1 and K-pair {0,1} / {2,3}
  v8f acc0 = {};
  v8f acc1 = {};
  int base = a0;
  // main loop: base+4 <= a1 ==> atom <= base+3 < a1, no guards needed
  for (; base + 4 <= a1; base += 4) {
    acc0 = ewald_step<false>(q, r, base     + half, a1, kt, kx, ky, kz, acc0);
    acc1 = ewald_step<false>(q, r, base + 2 + half, a1, kt, kx, ky, kz, acc1);
  }
  for (; base < a1; base += 2)                 // <=2 guarded steps
    acc0 = ewald_step<true>(q, r, base + half, a1, kt, kx, ky, kz, acc0);

  // lanes 0..15 hold D rows 0..3 at column kt: SRe0, SRe1, SIm0, SIm1
  // lanes 16..31 hold rows 8..11 which are exactly zero (A rows >=4 are zero)
  float s0 = acc0[0] + acc1[0];
  float s1 = acc0[1] + acc1[1];
  float s2 = acc0[2] + acc1[2];
  float s3 = acc0[3] + acc1[3];
  float contrib = w * (s0*s0 + s1*s1 + s2*s2 + s3*s3);
  for (int m = 16; m >= 1; m >>= 1) contrib += __shfl_xor(contrib, m, 32);
  if (lane == 0) atomicAdd(ws + WS_SACC + b, contrib);
}

__global__ void ewald_fin(const float* __restrict__ ws, float* __restrict__ out, int B) {
  int b = threadIdx.x;
  if (b < B) {
    float pot = ws[WS_SACC + b] / ws[WS_VOL + b] - ws[WS_SELF + b] * SELF_COEF;
    out[b] = pot * NORM_FACTOR;
  }
}

extern "C" void kernel_launch(void* const* d_in, const int* in_sizes, int n_in,
                              void* d_out, int out_size, void* d_ws, size_t ws_size,
                              hipStream_t stream) {
  const float* q    = (const float*)d_in[0];   // [N,2]
  const float* r    = (const float*)d_in[1];   // [N,3]
  const float* cell = (const float*)d_in[2];   // [B,3,3]
  const int*   batch= (const int*)d_in[3];     // [N] sorted
  float* ws  = (float*)d_ws;
  float* out = (float*)d_out;
  const int N = in_sizes[1] / 3;
  const int B = in_sizes[2] / 9;

  ewald_prep<<<1, 128, 0, stream>>>(q, cell, batch, ws, N, B);
  const int tiles  = B * TILES;
  const int blocks = (tiles + WAVES_PER_BLOCK - 1) / WAVES_PER_BLOCK;
  ewald_main<<<blocks, WAVES_PER_BLOCK * 32, 0, stream>>>(q, r, ws, B);
  ewald_fin<<<1, 32, 0, stream>>>(ws, out, B);
}